// DecodeWrapperEager_31516470018055
// MI455X (gfx1250) — compile-verified
//
#include <hip/hip_runtime.h>
#include <hip/hip_bf16.h>
#include <stdint.h>

typedef __attribute__((ext_vector_type(2))) float v2f;
typedef __attribute__((ext_vector_type(8))) float v8f;

#define USE_ASYNC_LDS 1

namespace {
constexpr int kB    = 8;
constexpr int kHq   = 32;
constexpr int kHk   = 8;
constexpr int kG    = 4;      // Hq / Hk
constexpr int kD    = 64;
constexpr int kS    = 32;     // block_seq_stride
constexpr int kNB   = 128;
constexpr int kW    = 1024;   // sliding window
constexpr int kNWave = 8;     // waves per workgroup
constexpr int kSplit = 8;     // workgroup splits per (b,hk)
constexpr int kPad  = 68;     // padded LDS row stride (floats) -> conflict-free gathers
constexpr int kTileFloats  = 16 * kPad;               // one 16x64 tile, padded = 1088
constexpr int kBufFloats   = 2 * kTileFloats;         // K+V tile              = 2176
constexpr int kWaveFloats  = 2 * kBufFloats;          // double buffered       = 4352
constexpr int kStageFloats = kNWave * kWaveFloats;    //                       = 34816
constexpr int kSmemFloats  = kStageFloats + kNWave * 4 /*m*/ +
                             kNWave * 4 /*l*/ + kNWave * 4 * kD /*acc*/;
constexpr int kPartStride  = 4 /*m*/ + 4 /*l*/ + 4 * kD /*acc*/;   // 264 floats per wg partial
}

__global__ __launch_bounds__(256)
void paged_decode_attn_split(const float* __restrict__ q_last,     // [B,1,Hq,D]
                             const float* __restrict__ k_last,     // [B,1,Hk,D]
                             const float* __restrict__ v_last,     // [B,1,Hk,D]
                             const float* __restrict__ cache,      // [P,2,Hk,S,D]
                             const int*   __restrict__ blk_ids,    // [B,NB]
                             const int*   __restrict__ start_pos,  // [B]
                             float*       __restrict__ ws)         // [B*Hk*Split, 264]
{
    __shared__ float smem[kSmemFloats];

    const int tid  = threadIdx.x;
    const int wave = tid >> 5;
    const int lane = tid & 31;
    const int n    = lane & 15;          // N column within WMMA tiles
    const int h    = lane >> 4;          // lane half
    const int koff = 2 * h;              // K offset inside 4-wide WMMA K dim

    const int bh    = blockIdx.x / kSplit;       // (b,hk) index
    const int split = blockIdx.x % kSplit;
    const int b  = bh / kHk;
    const int hk = bh % kHk;

    const int pos = start_pos[b];
    const int lo  = (pos - kW + 1) > 0 ? (pos - kW + 1) : 0;
    const int ti_lo = lo >> 4;
    const int ti_hi = pos >> 4;
    const int worker = split * kNWave + wave;    // 0..63 global worker per (b,hk)
    const int stride = kSplit * kNWave;          // 64

    // ---- Q^T B-operands, resident in registers (B[k][n'] = Q[g=n'][4i+k]) ----
    v2f qb[16];
    {
        const float* qrow = q_last + ((size_t)b * kHq + hk * kG + n) * kD;
        for (int i = 0; i < 16; ++i) {
            if (n < kG) {
                qb[i].x = qrow[4 * i + koff];
                qb[i].y = qrow[4 * i + koff + 1];
            } else {
                qb[i].x = 0.0f; qb[i].y = 0.0f;
            }
        }
    }

    // flash-attention running state (per lane; column n == q-group g)
    v8f accd[4];                             // out^T chunks: row d = 16c + 8h + r, col g = n
    for (int c = 0; c < 4; ++c)
        for (int r = 0; r < 8; ++r) accd[c][r] = 0.0f;
    float m_run = -1e30f;
    float l_run = 0.0f;

    float* const stage = smem + wave * kWaveFloats;
#if USE_ASYNC_LDS
    const uint32_t stageOff = (uint32_t)(uintptr_t)(void*)stage;
#endif

    // Issue one tile's K/V staging (async copy to LDS, or sync fallback).
    auto issue_tile = [&](int ti, int buf) {
        const int t0 = ti << 4;
        const int page = blk_ids[b * kNB + (t0 >> 5)];
        const int s0 = t0 & (kS - 1);
        const float* kg = cache + ((((size_t)page * 2 + 0) * kHk + hk) * kS + s0) * kD;
        const float* vg = cache + ((((size_t)page * 2 + 1) * kHk + hk) * kS + s0) * kD;
#if USE_ASYNC_LDS
        const uint32_t kOff = stageOff + (uint32_t)(buf * kBufFloats) * 4u;
        const uint32_t vOff = kOff + (uint32_t)kTileFloats * 4u;
        for (int it = 0; it < 8; ++it) {
            const int row = it * 2 + h;
            const uint32_t dst = (uint32_t)(row * kPad + 4 * n) * 4u;
            asm volatile("global_load_async_to_lds_b128 %0, %1, off"
                         :: "v"(kOff + dst),
                            "v"((uint64_t)(uintptr_t)(kg + row * kD + 4 * n))
                         : "memory");
            asm volatile("global_load_async_to_lds_b128 %0, %1, off"
                         :: "v"(vOff + dst),
                            "v"((uint64_t)(uintptr_t)(vg + row * kD + 4 * n))
                         : "memory");
        }
#else
        float* ksh = stage + buf * kBufFloats;
        float* vsh = ksh + kTileFloats;
        for (int it = 0; it < 8; ++it) {
            const int row = it * 2 + h;
            const float4 k4 = *(const float4*)(kg + row * kD + 4 * n);
            const float4 v4 = *(const float4*)(vg + row * kD + 4 * n);
            *(float4*)(ksh + row * kPad + 4 * n) = k4;
            *(float4*)(vsh + row * kPad + 4 * n) = v4;
        }
        // prefetch the tile after next while we compute (global_prefetch_b8)
        __builtin_prefetch(kg + (size_t)stride * 16 * kD + lane * 32, 0, 1);
        __builtin_prefetch(vg + (size_t)stride * 16 * kD + lane * 32, 0, 1);
#endif
    };

    int ti = ti_lo + worker;
    bool have = (ti <= ti_hi);
    int buf = 0;
    if (have) issue_tile(ti, buf);

    while (have) {
        const int tin = ti + stride;
        const bool have_next = (tin <= ti_hi);
        if (have_next) issue_tile(tin, buf ^ 1);
#if USE_ASYNC_LDS
        if (have_next) asm volatile("s_wait_asynccnt 0x10" ::: "memory");
        else           asm volatile("s_wait_asynccnt 0x0"  ::: "memory");
#endif
        float* ksh = stage + buf * kBufFloats;
        float* vsh = ksh + kTileFloats;
        const int t0 = ti << 4;

        // substitute the brand-new token's k/v (reference scatters it into cache)
        if (pos >= t0 && pos < t0 + 16) {
            const int row = pos - t0;
            const size_t kvoff = ((size_t)b * kHk + hk) * kD + 2 * lane;
            ksh[row * kPad + 2 * lane]     = k_last[kvoff];
            ksh[row * kPad + 2 * lane + 1] = k_last[kvoff + 1];
            vsh[row * kPad + 2 * lane]     = v_last[kvoff];
            vsh[row * kPad + 2 * lane + 1] = v_last[kvoff + 1];
        }

        // ---- S^T = K_tile x Q^T : 16 WMMA f32 16x16x4 steps over D ----
        v8f sc;
        for (int r = 0; r < 8; ++r) sc[r] = 0.0f;
        for (int i = 0; i < 16; ++i) {
            v2f a;
            a.x = ksh[n * kPad + 4 * i + koff];
            a.y = ksh[n * kPad + 4 * i + koff + 1];
            sc = __builtin_amdgcn_wmma_f32_16x16x4_f32(
                    false, a, false, qb[i], (short)0, sc, false, false);
        }

        // ---- mask + online softmax (column n == g; rows are positions) ----
        float s[8];
        float mloc = -1e30f;
        for (int r = 0; r < 8; ++r) {
            const int t = t0 + r + 8 * h;
            const float val = (t >= lo && t <= pos) ? sc[r] * 0.125f : -1e30f;
            s[r] = val;
            mloc = fmaxf(mloc, val);
        }
        mloc = fmaxf(mloc, __shfl_xor(mloc, 16, 32));
        const float mnew = fmaxf(m_run, mloc);
        const float rescale = __expf(m_run - mnew);
        float p[8];
        float ssum = 0.0f;
        for (int r = 0; r < 8; ++r) {
            p[r] = __expf(s[r] - mnew);
            ssum += p[r];
        }
        ssum += __shfl_xor(ssum, 16, 32);
        l_run = l_run * rescale + ssum;
        m_run = mnew;
        for (int c = 0; c < 4; ++c)
            for (int r = 0; r < 8; ++r) accd[c][r] *= rescale;

        // ---- build P^T B-operands from score-C layout (8 half-swaps) ----
        float psw[8];
        for (int r = 0; r < 8; ++r) psw[r] = __shfl_xor(p[r], 16, 32);
        const bool hi = (h != 0);
        v2f pb[4];
        pb[0].x = hi ? psw[2] : p[0];  pb[0].y = hi ? psw[3] : p[1];
        pb[1].x = hi ? psw[6] : p[4];  pb[1].y = hi ? psw[7] : p[5];
        pb[2].x = hi ? p[2] : psw[0];  pb[2].y = hi ? p[3] : psw[1];
        pb[3].x = hi ? p[6] : psw[4];  pb[3].y = hi ? p[7] : psw[5];

        // ---- out^T += V^T x P^T : 4 d-chunks x 4 k-steps of WMMA ----
        for (int c = 0; c < 4; ++c) {
            for (int kb = 0; kb < 4; ++kb) {
                const int t = kb * 4 + koff;
                v2f a;
                a.x = vsh[t * kPad + c * 16 + n];
                a.y = vsh[(t + 1) * kPad + c * 16 + n];
                accd[c] = __builtin_amdgcn_wmma_f32_16x16x4_f32(
                            false, a, false, pb[kb], (short)0, accd[c], false, false);
            }
        }

        ti = tin;
        buf ^= 1;
        have = have_next;
    }

    // ---- per-wave partials -> LDS ----
    float* pm = smem + kStageFloats;
    float* pl = pm + kNWave * 4;
    float* pa = pl + kNWave * 4;
    if (n < kG && h == 0) {
        pm[wave * 4 + n] = m_run;
        pl[wave * 4 + n] = l_run;
    }
    if (n < kG) {
        for (int c = 0; c < 4; ++c)
            for (int r = 0; r < 8; ++r)
                pa[(wave * 4 + n) * kD + c * 16 + 8 * h + r] = accd[c][r];
    }
    __syncthreads();

    // ---- merge this workgroup's 8 wave-partials -> workspace partial ----
    {
        const int g = tid >> 6;
        const int d = tid & 63;
        float M = -1e30f;
        for (int w = 0; w < kNWave; ++w) M = fmaxf(M, pm[w * 4 + g]);
        float L = 0.0f;
        float o = 0.0f;
        for (int w = 0; w < kNWave; ++w) {
            const float e = __expf(pm[w * 4 + g] - M);
            L += pl[w * 4 + g] * e;
            o += pa[(w * 4 + g) * kD + d] * e;
        }
        float* wsp = ws + (size_t)blockIdx.x * kPartStride;
        if (d == 0) {
            wsp[g]     = M;
            wsp[4 + g] = L;
        }
        wsp[8 + g * kD + d] = o;
    }
}

__global__ __launch_bounds__(256)
void merge_partials(const float* __restrict__ ws,     // [B*Hk*Split, 264]
                    const float* __restrict__ sink,   // [Hq]
                    float*       __restrict__ out)    // [B,1,Hq,D]
{
    const int bh = blockIdx.x;            // (b,hk)
    const int b  = bh / kHk;
    const int hk = bh % kHk;
    const int g  = threadIdx.x >> 6;
    const int d  = threadIdx.x & 63;

    float M = -1e30f;
    for (int sp = 0; sp < kSplit; ++sp)
        M = fmaxf(M, ws[(size_t)(bh * kSplit + sp) * kPartStride + g]);
    const float sk = sink[hk * kG + g];
    const float Mf = fmaxf(M, sk);
    float L = __expf(sk - Mf);            // sink mass (denominator only)
    float o = 0.0f;
    for (int sp = 0; sp < kSplit; ++sp) {
        const float* wsp = ws + (size_t)(bh * kSplit + sp) * kPartStride;
        const float e = __expf(wsp[g] - Mf);
        L += wsp[4 + g] * e;
        o += wsp[8 + g * kD + d] * e;
    }
    out[((size_t)b * kHq + hk * kG + g) * kD + d] = o / L;
}

extern "C" void kernel_launch(void* const* d_in, const int* in_sizes, int n_in,
                              void* d_out, int out_size, void* d_ws, size_t ws_size,
                              hipStream_t stream) {
    const float* q_last   = (const float*)d_in[0];
    const float* k_last   = (const float*)d_in[1];
    const float* v_last   = (const float*)d_in[2];
    const float* cache    = (const float*)d_in[3];
    const int*   blk_ids  = (const int*)d_in[4];
    const int*   startpos = (const int*)d_in[5];
    const float* sink     = (const float*)d_in[6];
    float* out = (float*)d_out;
    float* ws  = (float*)d_ws;
    (void)in_sizes; (void)n_in; (void)out_size; (void)ws_size;

    paged_decode_attn_split<<<dim3(kB * kHk * kSplit), dim3(256), 0, stream>>>(
        q_last, k_last, v_last, cache, blk_ids, startpos, ws);
    merge_partials<<<dim3(kB * kHk), dim3(256), 0, stream>>>(ws, sink, out);
}